// ProjectedGaussianRasterizer_7421703487873
// MI455X (gfx1250) — compile-verified
//
#include <hip/hip_runtime.h>
#include <hip/hip_bf16.h>
#include <math.h>

typedef __attribute__((ext_vector_type(2))) float v2f;
typedef __attribute__((ext_vector_type(8))) float v8f;

#define G 512
#define HH 224
#define WW 224
#define ALPHA_THR (1.0f / 255.0f)
#define ALPHA_CLAMP 0.9999f

// ---------------------------------------------------------------------------
// Kernel 1: bitonic depth sort (front-to-back) + pack sorted gaussians into
// workspace as three float4 SoA blocks:
//   ws[0..511]     = {mx, my, conic_a, conic_b}
//   ws[512..1023]  = {conic_c, opacity, cull_radius, 0}
//   ws[1024..1535] = {r, g, b, 0}
// cull_radius: alpha >= 1/255 requires sigma <= log(255*op); sigma >= 0.5*lam_min*|d|^2
// ---------------------------------------------------------------------------
__global__ __launch_bounds__(256) void gsplat_sort_pack(
    const float* __restrict__ means2d, const float* __restrict__ conics,
    const float* __restrict__ colors,  const float* __restrict__ opac,
    const float* __restrict__ depths,  float4* __restrict__ ws) {
  __shared__ float sk[G];
  __shared__ int   sv[G];
  const int t = threadIdx.x;
  for (int i = t; i < G; i += 256) { sk[i] = depths[i]; sv[i] = i; }
  __syncthreads();
  for (int k = 2; k <= G; k <<= 1) {
    for (int j = k >> 1; j > 0; j >>= 1) {
      for (int i = t; i < G; i += 256) {
        int ixj = i ^ j;
        if (ixj > i) {
          bool asc = ((i & k) == 0);
          float ki = sk[i], kj = sk[ixj];
          bool swap = asc ? (ki > kj) : (ki < kj);
          if (swap) {
            sk[i] = kj; sk[ixj] = ki;
            int vi = sv[i]; sv[i] = sv[ixj]; sv[ixj] = vi;
          }
        }
      }
      __syncthreads();
    }
  }
  for (int i = t; i < G; i += 256) {
    int s = sv[i];
    float mx = means2d[2 * s], my = means2d[2 * s + 1];
    float a = conics[3 * s], b = conics[3 * s + 1], c = conics[3 * s + 2];
    float o = opac[s];
    float tr = a + c, df = a - c;
    float lam_min = 0.5f * (tr - sqrtf(df * df + 4.0f * b * b));
    float cutoff = fmaxf(logf(255.0f * o), 0.0f);
    float rad = sqrtf(2.0f * cutoff / fmaxf(lam_min, 1e-12f)) + 1.0f;
    ws[i]         = make_float4(mx, my, a, b);
    ws[G + i]     = make_float4(c, o, rad, 0.0f);
    ws[2 * G + i] = make_float4(colors[3 * s], colors[3 * s + 1],
                                colors[3 * s + 2], 0.0f);
  }
}

__device__ __forceinline__ float alpha_of(float4 e, float4 f, float px, float py) {
  float dx = px - e.x, dy = py - e.y;
  float sigma = 0.5f * (e.z * dx * dx + f.x * dy * dy) + e.w * dx * dy;
  float al = fminf(f.y * __expf(-sigma), ALPHA_CLAMP);
  return (sigma >= 0.0f && al >= ALPHA_THR) ? al : 0.0f;
}

// ---------------------------------------------------------------------------
// Kernel 2: tiled rasterizer. 256 threads = 8 waves; each wave = a 16x1 pixel
// strip. Per chunk of 4 sorted gaussians:
//   - wave-uniform bbox cull (lane tests gaussian chunk + (lane&3))
//   - lane (pixel = lane&15, K-pair = 2*(lane>>4)) computes 2 alphas
//   - one shfl_xor(16) merges (1-a) products between K halves -> exact
//     exclusive transmittance weights in the WMMA A-matrix layout
//   - V_WMMA_F32_16X16X4_F32 accumulates weights x colors into C (N<3 = RGB)
//   - wave-uniform early exit when T < 1e-6 everywhere
// ---------------------------------------------------------------------------
__global__ __launch_bounds__(256) void gsplat_raster(
    const float4* __restrict__ ws, float* __restrict__ out) {
  __shared__ float4 sGe[G];
  __shared__ float4 sG2[G];
  __shared__ float  sCol[G * 4];
  const int t = threadIdx.x;
  for (int i = t; i < G; i += 256) {
    sGe[i] = ws[i];
    sG2[i] = ws[G + i];
    float4 c = ws[2 * G + i];
    sCol[4 * i + 0] = c.x; sCol[4 * i + 1] = c.y;
    sCol[4 * i + 2] = c.z; sCol[4 * i + 3] = 0.0f;
  }
  __syncthreads();

  const int lane = t & 31;
  const int wv   = t >> 5;
  const int x0   = blockIdx.x * 16;
  const int y    = blockIdx.y * 8 + wv;
  const int n    = lane & 15;         // A: pixel row M;  B: channel col N
  const int hi   = lane >> 4;         // K half: 0 -> K{0,1}, 1 -> K{2,3}
  const int kb   = hi << 1;
  const float px = (float)(x0 + n) + 0.5f;
  const float py = (float)y + 0.5f;
  const float rx0 = (float)x0 + 0.5f, rx1 = (float)x0 + 15.5f;
  const int   gt  = lane & 3;         // gaussian tested by this lane for cull
  const int   cidx = (n < 4) ? n : 3; // N>=3 reads the zero pad -> B col = 0

  float T = 1.0f;
  v8f acc = {};

  for (int ch = 0; ch < G; ch += 4) {
    // --- conservative strip cull (wave-uniform decision) ---
    float4 gg = sGe[ch + gt];
    float rad = sG2[ch + gt].z;
    float ddx = fmaxf(fmaxf(rx0 - gg.x, gg.x - rx1), 0.0f);
    float ddy = fabsf(py - gg.y);
    bool hit = (ddx * ddx + ddy * ddy) <= rad * rad;
    if (!__any(hit)) continue;

    // --- alphas for this lane's (pixel, K-pair) ---
    const int g0 = ch + kb;
    float a0 = alpha_of(sGe[g0],     sG2[g0],     px, py);
    float a1 = alpha_of(sGe[g0 + 1], sG2[g0 + 1], px, py);
    float om0 = 1.0f - a0;
    float m01 = om0 * (1.0f - a1);
    float mo  = __shfl_xor(m01, 16, 32);     // other K-half's (1-a)(1-a)
    float pre = hi ? mo : 1.0f;              // exclusive prefix across K
    float w0 = T * pre * a0;                 // A[M][K=kb]
    float w1 = T * pre * om0 * a1;           // A[M][K=kb+1]
    T *= m01 * mo;                           // same on both halves

    // --- WMMA accumulate: D = W(16x4) x Col(4x16) + C ---
    v2f A; A.x = w0; A.y = w1;
    v2f B; B.x = sCol[4 * g0 + cidx]; B.y = sCol[4 * (g0 + 1) + cidx];
    acc = __builtin_amdgcn_wmma_f32_16x16x4_f32(
        false, A, false, B, (short)0, acc, false, false);

    if (!__any(T > 1e-6f)) break;            // wave-uniform early exit
  }

  // C/D layout: element (VGPR r, lane) = (M = r + 8*hi, N = lane&15)
  if (n < 3) {
    const int baseM = hi * 8;
    #pragma unroll
    for (int r = 0; r < 8; ++r) {
      int xg = x0 + baseM + r;
      out[(y * WW + xg) * 3 + n] = acc[r];
    }
  }
}

extern "C" void kernel_launch(void* const* d_in, const int* in_sizes, int n_in,
                              void* d_out, int out_size, void* d_ws, size_t ws_size,
                              hipStream_t stream) {
  const float* means2d = (const float*)d_in[0];
  const float* conics  = (const float*)d_in[1];
  const float* colors  = (const float*)d_in[2];
  const float* opac    = (const float*)d_in[3];
  const float* depths  = (const float*)d_in[4];
  float* out = (float*)d_out;
  float4* ws = (float4*)d_ws;   // 3 * 512 * 16B = 24 KB

  gsplat_sort_pack<<<1, 256, 0, stream>>>(means2d, conics, colors, opac, depths, ws);
  gsplat_raster<<<dim3(WW / 16, HH / 8), 256, 0, stream>>>(ws, out);
}